// MMD_loss_7155415515714
// MI455X (gfx1250) — compile-verified
//
#include <hip/hip_runtime.h>
#include <hip/hip_bf16.h>

typedef float v2f __attribute__((ext_vector_type(2)));
typedef float v8f __attribute__((ext_vector_type(8)));

#define NTOT 8192
#define BATCH 4096
#define DIM 64
#define TILE 128
#define LDST 68  // padded LDS row stride (floats): bank = (4m + k) % 64, conflict-free

#if __has_builtin(__builtin_amdgcn_exp2f)
#define FAST_EXP2(x) __builtin_amdgcn_exp2f(x)
#else
#define FAST_EXP2(x) exp2f(x)
#endif

// ---------------- ws layout (floats) ----------------
// [0..8191]   sq[i]
// [8192]      sum of sq (accumulator)
// [8193]      sum of column-sum squares (accumulator)
// [8194]      nbase = -log2(e)/(16*bw)
// [8199]      loss accumulator

__global__ void mmd_init_kernel(float* ws) {
    ws[8192] = 0.0f;
    ws[8193] = 0.0f;
    ws[8199] = 0.0f;
}

// one thread per row: sq[row] = ||x_row||^2 ; block-reduce sum(sq) -> atomic
__global__ __launch_bounds__(256) void mmd_rowsq_kernel(
    const float* __restrict__ src, const float* __restrict__ tgt,
    float* __restrict__ sq, float* __restrict__ sumsq) {
    int row = blockIdx.x * 256 + threadIdx.x;
    const float* p = (row < BATCH) ? (src + row * DIM) : (tgt + (row - BATCH) * DIM);
    float s = 0.0f;
#pragma unroll
    for (int c = 0; c < DIM; c += 4) {
        float4 v = *(const float4*)(p + c);
        s += v.x * v.x + v.y * v.y + v.z * v.z + v.w * v.w;
    }
    sq[row] = s;
    __shared__ float red[256];
    red[threadIdx.x] = s;
    __syncthreads();
    for (int off = 128; off > 0; off >>= 1) {
        if (threadIdx.x < (unsigned)off) red[threadIdx.x] += red[threadIdx.x + off];
        __syncthreads();
    }
    if (threadIdx.x == 0) atomicAdd(sumsq, red[0]);
}

// one block per column c: s_c = sum_i x[i,c]; atomically accumulate s_c^2
__global__ __launch_bounds__(256) void mmd_colsum_kernel(
    const float* __restrict__ src, const float* __restrict__ tgt,
    float* __restrict__ s2) {
    int c = blockIdx.x;
    float s = 0.0f;
    for (int r = threadIdx.x; r < NTOT; r += 256) {
        const float* p = (r < BATCH) ? (src + r * DIM) : (tgt + (r - BATCH) * DIM);
        s += p[c];
    }
    __shared__ float red[256];
    red[threadIdx.x] = s;
    __syncthreads();
    for (int off = 128; off > 0; off >>= 1) {
        if (threadIdx.x < (unsigned)off) red[threadIdx.x] += red[threadIdx.x + off];
        __syncthreads();
    }
    if (threadIdx.x == 0) atomicAdd(s2, red[0] * red[0]);
}

// sum(L2) = 2N*sum(sq) - 2*sum_c(colsum_c^2); bw = sumL2/(N^2-N)/4.
// Geometric bandwidth ladder: sum_m exp(-L2/(bw*2^m)) = e + e^2 + e^4 + e^8 + e^16
// with e = exp(-L2/(16*bw)) = exp2(L2 * nbase), nbase = -log2(e)/(16*bw).
__global__ void mmd_bw_kernel(const float* __restrict__ sumsq,
                              const float* __restrict__ s2,
                              float* __restrict__ nbase) {
    float sumL2 = 2.0f * (float)NTOT * sumsq[0] - 2.0f * s2[0];
    float denom = (float)NTOT * (float)NTOT - (float)NTOT;
    float bw = sumL2 / denom * 0.25f;  // / KERNEL_MUL^(KERNEL_NUM//2) = /4
    nbase[0] = -1.44269504088896340736f / (16.0f * bw);
}

// fused tile kernel: 128x128 Gram tile via V_WMMA_F32_16X16X4_F32, then
// 1-exp geometric-ladder Gaussian kernel sum with quadrant sign.
// Symmetry: only tiles with by >= bx run; off-diagonal tiles weighted 2x.
__global__ __launch_bounds__(256) void mmd_tile_kernel(
    const float* __restrict__ src, const float* __restrict__ tgt,
    const float* __restrict__ sq, const float* __restrict__ nbase,
    float* __restrict__ loss) {
    if (blockIdx.x > blockIdx.y) return;  // triangular: K is symmetric

    __shared__ float As[TILE * LDST];
    __shared__ float Bs[TILE * LDST];
    __shared__ float sqI[TILE];
    __shared__ float sqJ[TILE];
    __shared__ float wred[8];

    const int i0 = blockIdx.y * TILE;
    const int j0 = blockIdx.x * TILE;
    const float* pi = (i0 < BATCH) ? (src + i0 * DIM) : (tgt + (i0 - BATCH) * DIM);
    const float* pj = (j0 < BATCH) ? (src + j0 * DIM) : (tgt + (j0 - BATCH) * DIM);
    const int tid = threadIdx.x;

    // stage tiles: 128 rows x 64 cols per matrix, float4 moves, padded LDS stride
#pragma unroll
    for (int t = 0; t < 8; ++t) {
        int idx = tid + t * 256;        // 0..2047
        int row = idx >> 4;             // 16 float4 per row
        int c4 = (idx & 15) << 2;
        *(float4*)&As[row * LDST + c4] = *(const float4*)(pi + row * DIM + c4);
        *(float4*)&Bs[row * LDST + c4] = *(const float4*)(pj + row * DIM + c4);
    }
    if (tid < TILE) sqI[tid] = sq[i0 + tid];
    else sqJ[tid - TILE] = sq[j0 + tid - TILE];
    __syncthreads();

    const int wave = tid >> 5;
    const int lane = tid & 31;
    const int mrow = lane & 15;       // row within 16-row frag
    const int hi = lane >> 4;         // K-half select
    const int kh = hi << 1;           // K offset {0,2}

    v8f zero = {};
    v8f acc[8];
#pragma unroll
    for (int s = 0; s < 8; ++s) acc[s] = zero;

    const float* aBase = &As[(wave * 16 + mrow) * LDST + kh];

#pragma unroll
    for (int k0 = 0; k0 < DIM; k0 += 4) {
        v2f a = *(const v2f*)(aBase + k0);
#pragma unroll
        for (int s = 0; s < 8; ++s) {
            v2f b = *(const v2f*)&Bs[(s * 16 + mrow) * LDST + kh + k0];
            acc[s] = __builtin_amdgcn_wmma_f32_16x16x4_f32(
                false, a, false, b, (short)0, acc[s], false, false);
        }
    }

    // epilogue: e = exp2(L2*nbase); kernel sum = e + e^2 + e^4 + e^8 + e^16
    const float nb = nbase[0];
    const int ncol = lane & 15;
    float partial = 0.0f;
#pragma unroll
    for (int s = 0; s < 8; ++s) {
        float sj = sqJ[s * 16 + ncol];
#pragma unroll
        for (int r = 0; r < 8; ++r) {
            int m = wave * 16 + r + 8 * hi;  // C/D layout: lanes16-31 hold M=r+8
            float L2 = sqI[m] + sj - 2.0f * acc[s][r];
            float e1 = FAST_EXP2(L2 * nb);   // exp(-L2/(16bw))
            float e2 = e1 * e1;              // /(8bw)
            float e4 = e2 * e2;              // /(4bw)
            float e8 = e4 * e4;              // /(2bw)
            float e16 = e8 * e8;             // /bw
            partial += e1 + e2 + e4 + e8 + e16;
        }
    }
    const float sign = ((i0 < BATCH) == (j0 < BATCH)) ? 1.0f : -1.0f;
    const float weight = (blockIdx.x == blockIdx.y) ? 1.0f : 2.0f;

#pragma unroll
    for (int off = 16; off > 0; off >>= 1)
        partial += __shfl_down(partial, (unsigned)off, 32);
    if (lane == 0) wred[wave] = partial;
    __syncthreads();
    if (tid == 0) {
        float t = 0.0f;
#pragma unroll
        for (int w = 0; w < 8; ++w) t += wred[w];
        atomicAdd(loss, sign * weight * t);
    }
}

__global__ void mmd_final_kernel(const float* __restrict__ loss, float* __restrict__ out) {
    out[0] = loss[0] * (1.0f / ((float)BATCH * (float)BATCH));
}

extern "C" void kernel_launch(void* const* d_in, const int* in_sizes, int n_in,
                              void* d_out, int out_size, void* d_ws, size_t ws_size,
                              hipStream_t stream) {
    const float* src = (const float*)d_in[0];
    const float* tgt = (const float*)d_in[1];
    float* ws = (float*)d_ws;
    float* sq = ws;             // 8192
    float* sumsq = ws + 8192;   // 1
    float* s2 = ws + 8193;      // 1
    float* nbase = ws + 8194;   // 1
    float* loss = ws + 8199;    // 1
    float* out = (float*)d_out;

    mmd_init_kernel<<<1, 1, 0, stream>>>(ws);
    mmd_rowsq_kernel<<<NTOT / 256, 256, 0, stream>>>(src, tgt, sq, sumsq);
    mmd_colsum_kernel<<<DIM, 256, 0, stream>>>(src, tgt, s2);
    mmd_bw_kernel<<<1, 1, 0, stream>>>(sumsq, s2, nbase);
    dim3 grid(NTOT / TILE, NTOT / TILE);
    mmd_tile_kernel<<<grid, 256, 0, stream>>>(src, tgt, sq, nbase, loss);
    mmd_final_kernel<<<1, 1, 0, stream>>>(loss, out);
}